// RadonTransform_40673340293404
// MI455X (gfx1250) — compile-verified
//
#include <hip/hip_runtime.h>
#include <hip/hip_bf16.h>

typedef float v2f __attribute__((ext_vector_type(2)));
typedef float v8f __attribute__((ext_vector_type(8)));

#define IMG_H 512
#define IMG_W 512
#define N_ANG 256

// One zero-padded bilinear tap (clamped load + mask keeps EXEC all-ones).
__device__ __forceinline__ float radon_tap(const float* __restrict__ img, int y, int x) {
    unsigned ux = (unsigned)x;
    unsigned uy = (unsigned)y;
    bool valid = (ux < (unsigned)IMG_W) && (uy < (unsigned)IMG_H);
    unsigned xc = ux < (unsigned)IMG_W ? ux : (unsigned)(IMG_W - 1);
    unsigned yc = uy < (unsigned)IMG_H ? uy : (unsigned)(IMG_H - 1);
    float v = img[yc * IMG_W + xc];
    return valid ? v : 0.0f;
}

// grid_sample(align_corners=False, zero padding) bilinear sample of the image
// at the rotated coordinate of (cx, cy), both centered (i.e. x - 255.5).
__device__ __forceinline__ float radon_sample(const float* __restrict__ img,
                                              float c, float s,
                                              float cx, float cy) {
    float ix = fmaf(s, cy, fmaf(c, cx, 255.5f));   //  cos*cx + sin*cy + 255.5
    float iy = fmaf(c, cy, fmaf(-s, cx, 255.5f));  // -sin*cx + cos*cy + 255.5
    float fx = floorf(ix);
    float fy = floorf(iy);
    int x0 = (int)fx;
    int y0 = (int)fy;
    float wx1 = ix - fx;
    float wy1 = iy - fy;
    float wx0 = 1.0f - wx1;
    float wy0 = 1.0f - wy1;
    float v00 = radon_tap(img, y0,     x0);
    float v01 = radon_tap(img, y0,     x0 + 1);
    float v10 = radon_tap(img, y0 + 1, x0);
    float v11 = radon_tap(img, y0 + 1, x0 + 1);
    float r0 = fmaf(wx0, v00, wx1 * v01);
    float r1 = fmaf(wx0, v10, wx1 * v11);
    return fmaf(wy0, r0, wy1 * r1);
}

// One wave32 per (angle, 16-wide x tile). Lanes 0-15 sample rows {y, y+1} for
// column x = xbase+lane; lanes 16-31 sample rows {y+2, y+3}. A single
// V_WMMA_F32_16X16X4_F32 with an all-ones A matrix reduces those 4 rows x 16
// columns into the fp32 accumulator; 128 WMMAs sweep all 512 rows. Since every
// row of A is 1.0, D[m][n] = sum_k B[k][n] regardless of the K<->lane mapping,
// and all 16 rows of D hold the projection — lanes 0-15 read D VGPR0.
__global__ void __launch_bounds__(256)
radon_wmma_kernel(const float* __restrict__ img,
                  const float* __restrict__ angles,
                  float* __restrict__ out) {
    const int lane  = threadIdx.x & 31;
    const int wave  = (blockIdx.x << 3) + (threadIdx.x >> 5);  // 8 waves / block
    const int ang   = wave >> 5;         // 32 x-tiles per angle
    const int tile  = wave & 31;
    const int xbase = tile << 4;

    const float a = angles[ang];
    const float c = cosf(a);
    const float s = sinf(a);

    const int n    = lane & 15;          // column within the tile
    const int ksub = (lane >> 4) << 1;   // 0 for lanes 0-15, 2 for lanes 16-31

    const float cx = (float)(xbase + n) - 255.5f;

    const v2f ones = {1.0f, 1.0f};       // A matrix: all ones (both K slots)
    v8f acc = {};                        // C/D accumulator, fp32

    #pragma unroll 4
    for (int y = 0; y < IMG_H; y += 4) {
        float cy = (float)(y + ksub) - 255.5f;
        v2f b;
        b.x = radon_sample(img, c, s, cx, cy);
        b.y = radon_sample(img, c, s, cx, cy + 1.0f);
        // D = ones(16x4) * B(4x16) + C  ->  row-sum accumulate
        acc = __builtin_amdgcn_wmma_f32_16x16x4_f32(
            /*neg_a=*/false, ones,
            /*neg_b=*/false, b,
            /*c_mod=*/(short)0, acc,
            /*reuse_a=*/false, /*reuse_b=*/false);
    }

    if (lane < 16) {
        out[ang * IMG_W + xbase + n] = acc[0];
    }
}

extern "C" void kernel_launch(void* const* d_in, const int* in_sizes, int n_in,
                              void* d_out, int out_size, void* d_ws, size_t ws_size,
                              hipStream_t stream) {
    const float* img    = (const float*)d_in[0];   // [1,1,512,512] fp32
    const float* angles = (const float*)d_in[1];   // [256] fp32
    float* out          = (float*)d_out;           // [1,1,256,512] fp32

    // 256 angles * 32 x-tiles = 8192 waves; 8 waves (256 threads) per block.
    dim3 grid(1024);
    dim3 block(256);
    radon_wmma_kernel<<<grid, block, 0, stream>>>(img, angles, out);
}